// BTTLayer_36885179138559
// MI455X (gfx1250) — compile-verified
//
#include <hip/hip_runtime.h>
#include <hip/hip_bf16.h>

// BTT layer fused kernel for gfx1250 (MI455X).
//   x     : (4096, 4096) fp32   [B rows, n*64+b cols]
//   btt_r : (64, 64, 512) fp32  [n, b, m*8+r]
//   btt_l : (64, 512, 64) fp32  [m, n*8+r, a]
//   out   : (4096, 4096) fp32   [B rows, m*64+a]
//
// out[B, m*64+a] = sum_n ( sum_b x[B,n*64+b] * btt_r[n,b,m*8+r] ) * btt_l[m,n*8+r,a]
//
// Workgroup = 256 threads (8 wave32). Each WG: 128 B-rows x one m-pair (m0, m0+1).
// Each wave owns 16 B-rows; accumulates 16x128 of output in 8 v8f WMMA accumulators.

typedef __attribute__((ext_vector_type(2))) float v2f;
typedef __attribute__((ext_vector_type(8))) float v8f;

#define WMMA_F32(A, B, C) \
    __builtin_amdgcn_wmma_f32_16x16x4_f32(false, (A), false, (B), (short)0, (C), false, false)

__global__ __launch_bounds__(256) void btt_fused_kernel(
    const float* __restrict__ x,
    const float* __restrict__ btt_r,
    const float* __restrict__ btt_l,
    const float* __restrict__ bias,
    float* __restrict__ out)
{
    // LDS staging: R slice (64 b x 16 j), L rows (2 m-halves x 8 k x 64 a),
    // per-wave 16x16 T relayout scratch.
    __shared__ alignas(16) float sR[64 * 16];      // 4 KB
    __shared__ alignas(16) float sL[2 * 8 * 64];   // 4 KB
    __shared__ alignas(16) float sT[8 * 256];      // 8 KB (per-wave 1 KB)

    const int tid  = threadIdx.x;
    const int wave = tid >> 5;
    const int lane = tid & 31;
    const int lo   = lane & 15;   // lane % 16
    const int hi   = lane >> 4;   // 0 for lanes 0-15, 1 for lanes 16-31

    const int m0    = blockIdx.y * 2;                     // m-pair base
    const int rowg  = blockIdx.x * 128 + wave * 16 + lo;  // this lane's B-row (A operand)
    const float* xrow = x + (size_t)rowg * 4096;

    // Pre-decompose staging indices (uniform per thread, no divergence).
    const int st_b  = tid >> 2;          // 0..63   (R: b index)
    const int st_j4 = (tid & 3) << 2;    // 0,4,8,12 (R: j group of 4)
    const int st_h  = tid >> 7;          // 0..1    (L: m-half)
    const int st_k  = (tid & 127) >> 4;  // 0..7    (L: k row)
    const int st_c4 = (tid & 15) << 2;   // 0..60   (L: a group of 4)

    v8f acc[2][4] = {};  // [m-half][a-tile of 16] -> 16x16 f32 each

    for (int n = 0; n < 64; ++n) {
        __syncthreads();  // previous iteration's LDS reads complete

        // ---- cooperative stage of weight slices (coalesced float4) ----
        {
            const float4 gr = *reinterpret_cast<const float4*>(
                btt_r + (size_t)(n * 64 + st_b) * 512 + m0 * 8 + st_j4);
            *reinterpret_cast<float4*>(&sR[st_b * 16 + st_j4]) = gr;

            const float4 gl = *reinterpret_cast<const float4*>(
                btt_l + (size_t)((m0 + st_h) * 512 + n * 8 + st_k) * 64 + st_c4);
            *reinterpret_cast<float4*>(&sL[(st_h * 8 + st_k) * 64 + st_c4]) = gl;
        }
        __syncthreads();

        // ---- GEMM1: T(16x16) = X_strip(16x64) @ [R_n,m0 | R_n,m0+1](64x16) ----
        v8f t = {};
        const float* xn = xrow + n * 64;
        #pragma unroll
        for (int kk = 0; kk < 16; ++kk) {
            const int k0 = kk * 4;
            // A 16x4 layout: lanes0-15 hold K=k0,k0+1; lanes16-31 hold K=k0+2,k0+3
            v2f a = *reinterpret_cast<const v2f*>(xn + k0 + 2 * hi);
            // B 4x16 layout: VGPR0 rows K=k0 / k0+2, VGPR1 rows K=k0+1 / k0+3
            v2f b;
            b[0] = sR[(k0 + 2 * hi) * 16 + lo];
            b[1] = sR[(k0 + 2 * hi + 1) * 16 + lo];
            t = WMMA_F32(a, b, t);
        }

        // ---- spill T (C layout) to per-wave LDS scratch, re-read in A layout ----
        float* tw = &sT[wave * 256];
        #pragma unroll
        for (int v = 0; v < 8; ++v)
            tw[(v + 8 * hi) * 16 + lo] = t[v];   // row = v + 8*hi, col = lo
        asm volatile("s_wait_dscnt 0" ::: "memory");  // stores visible wave-wide

        // ---- GEMM2: acc[h] += T[:, h*8:+8](16x8) @ L[m0+h, n*8:+8, :](8x64) ----
        #pragma unroll
        for (int h = 0; h < 2; ++h) {
            #pragma unroll
            for (int c = 0; c < 2; ++c) {          // K chunks of 4 over r=0..7
                const int kb = c * 4 + 2 * hi;
                v2f a2;
                a2[0] = tw[lo * 16 + h * 8 + kb];
                a2[1] = tw[lo * 16 + h * 8 + kb + 1];
                #pragma unroll
                for (int j = 0; j < 4; ++j) {      // 4 a-tiles of 16
                    v2f b2;
                    b2[0] = sL[(h * 8 + kb) * 64 + j * 16 + lo];
                    b2[1] = sL[(h * 8 + kb + 1) * 64 + j * 16 + lo];
                    acc[h][j] = WMMA_F32(a2, b2, acc[h][j]);
                }
            }
        }
    }

    // ---- epilogue: C layout scatter + bias ----
    #pragma unroll
    for (int h = 0; h < 2; ++h) {
        #pragma unroll
        for (int j = 0; j < 4; ++j) {
            const int col = (m0 + h) * 64 + j * 16 + lo;
            const float bv = bias[col];
            #pragma unroll
            for (int v = 0; v < 8; ++v) {
                const int row = blockIdx.x * 128 + wave * 16 + v + 8 * hi;
                out[(size_t)row * 4096 + col] = acc[h][j][v] + bv;
            }
        }
    }
}

extern "C" void kernel_launch(void* const* d_in, const int* in_sizes, int n_in,
                              void* d_out, int out_size, void* d_ws, size_t ws_size,
                              hipStream_t stream) {
    const float* x     = (const float*)d_in[0];
    const float* btt_r = (const float*)d_in[1];
    const float* btt_l = (const float*)d_in[2];
    const float* bias  = (const float*)d_in[3];
    float* out = (float*)d_out;

    dim3 grid(32, 32, 1);   // 32 B-tiles of 128 rows x 32 m-pairs
    dim3 block(256, 1, 1);  // 8 wave32
    btt_fused_kernel<<<grid, block, 0, stream>>>(x, btt_r, btt_l, bias, out);
}